// GPTransBlock_11192684773579
// MI455X (gfx1250) — compile-verified
//
#include <hip/hip_runtime.h>
#include <math.h>

// GPTrans block for MI455X (gfx1250), fp32 WMMA (V_WMMA_F32_16X16X4_F32)
// + Tensor Data Mover staging of the strided [32 x 512] channel slabs.
// B=8 N=512 C=512 E=32 H=32 HD=16. Memory-bound on [B,{E,H},N,N] planes.

#define Bb 8
#define Nn 512
#define Cc 512
#define Ee 32
#define Hh 32
#define HDd 16
static constexpr float LN_EPS = 1e-5f;
static constexpr float SCALE = 0.25f;  // HD^-0.5
static constexpr size_t NNp = (size_t)Nn * Nn;

typedef float v2f __attribute__((ext_vector_type(2)));
typedef float v8f __attribute__((ext_vector_type(8)));
typedef unsigned uv4 __attribute__((ext_vector_type(4)));
typedef unsigned uv8 __attribute__((ext_vector_type(8)));

__device__ __forceinline__ v8f wmma4(v2f a, v2f b, v8f c) {
  // D = A(16x4,f32) * B(4x16,f32) + C(16x16,f32)
  return __builtin_amdgcn_wmma_f32_16x16x4_f32(false, a, false, b, (short)0, c,
                                               false, false);
}

__device__ __forceinline__ float wave_sum(float v) {
#pragma unroll
  for (int off = 16; off > 0; off >>= 1) v += __shfl_xor(v, off);
  return v;
}
__device__ __forceinline__ float wave_max(float v) {
#pragma unroll
  for (int off = 16; off > 0; off >>= 1) v = fmaxf(v, __shfl_xor(v, off));
  return v;
}
__device__ __forceinline__ float gelu_exact(float x) {
  return 0.5f * x * (1.0f + erff(x * 0.70710678118654752440f));
}

// --- Tensor Data Mover: 2-D tile (dim0 contiguous f32, dim1 strided) -> LDS.
// D# per CDNA5 ISA: group0 (count/lds/global/type),
// group1 (data_size=4B, tensor+tile dims, dim0 stride). 2-D => VADDR2/3 NULL.
__device__ __forceinline__ void tdm_load_2d_f32(const float* gptr, void* ldsptr,
                                                unsigned dim0, unsigned dim1,
                                                unsigned stride0_elems) {
  const unsigned long long ga = (unsigned long long)(size_t)gptr;
  const unsigned lds_off = (unsigned)(size_t)ldsptr;  // low 32b of flat = LDS byte offset
  uv4 g0;
  g0[0] = 1u;                                            // count=1, no gather
  g0[1] = lds_off;                                       // lds_addr
  g0[2] = (unsigned)(ga & 0xFFFFFFFFu);                  // global_addr[31:0]
  g0[3] = (unsigned)((ga >> 32) & 0x1FFFFFFu) | (2u << 30);  // addr[56:32] | type=2
  uv8 g1;
  g1[0] = (2u << 16);                                    // wg_mask=0, data_size=4B
  g1[1] = (dim0 & 0xFFFFu) << 16;                        // tensor_dim0 lo16
  g1[2] = ((dim0 >> 16) & 0xFFFFu) | ((dim1 & 0xFFFFu) << 16);  // dim0 hi | dim1 lo
  g1[3] = ((dim1 >> 16) & 0xFFFFu) | ((dim0 & 0xFFFFu) << 16);  // dim1 hi | tile_dim0
  g1[4] = dim1 & 0xFFFFu;                                // tile_dim1 (tile_dim2=0)
  g1[5] = stride0_elems;                                 // tensor_dim0_stride lo32
  g1[6] = 0u;
  g1[7] = 0u;
  asm volatile("tensor_load_to_lds %0, %1" ::"s"(g0), "s"(g1) : "memory");
}

// ---------------- LayerNorm: one block per row of C=512 -------------------
__global__ void ln_kernel(const float* __restrict__ x, const float* __restrict__ g,
                          const float* __restrict__ b, float* __restrict__ out) {
  const int row = blockIdx.x;  // B*N rows
  const float* xr = x + (size_t)row * Cc;
  float* orow = out + (size_t)row * Cc;
  const int t = threadIdx.x;  // 256 threads, 2 elems each
  float v0 = xr[t], v1 = xr[t + 256];
  float s = v0 + v1, ss = v0 * v0 + v1 * v1;
  __shared__ float red[18];
  s = wave_sum(s);
  ss = wave_sum(ss);
  const int wid = t >> 5;
  if ((t & 31) == 0) { red[wid] = s; red[wid + 8] = ss; }
  __syncthreads();
  if (t == 0) {
    float S = 0.f, SS = 0.f;
#pragma unroll
    for (int i = 0; i < 8; ++i) { S += red[i]; SS += red[i + 8]; }
    red[16] = S; red[17] = SS;
  }
  __syncthreads();
  const float mean = red[16] * (1.0f / Cc);
  const float var = red[17] * (1.0f / Cc) - mean * mean;
  const float inv = rsqrtf(var + LN_EPS);
  orow[t] = (v0 - mean) * inv * g[t] + b[t];
  orow[t + 256] = (v1 - mean) * inv * g[t + 256] + b[t + 256];
}

// ------------- Generic WMMA GEMM: Out = act(A @ W^T + bias) [+ Cadd] ------
// A: [rows, K], W: [cols, K] row-major, Out/Cadd: ldo. One wave = 16x64 tile.
template <int ACT>
__global__ __launch_bounds__(32) void gemm_awt_kernel(
    const float* __restrict__ A, const float* __restrict__ W,
    const float* __restrict__ bias, const float* __restrict__ Cadd,
    float* __restrict__ Out, int K, int ldo) {
  const int row0 = blockIdx.x * 16, col0 = blockIdx.y * 64;
  const int lane = threadIdx.x & 31, half = lane >> 4, l16 = lane & 15;
  v8f c[4];
#pragma unroll
  for (int t = 0; t < 4; ++t) {
    const float binit = bias[col0 + t * 16 + l16];
#pragma unroll
    for (int r = 0; r < 8; ++r) c[t][r] = binit;
  }
  const float* arow = A + (size_t)(row0 + l16) * K;
  for (int k = 0; k < K; k += 4) {
    v2f av;
    av.x = arow[k + 2 * half];
    av.y = arow[k + 2 * half + 1];
#pragma unroll
    for (int t = 0; t < 4; ++t) {
      const float* wr = W + (size_t)(col0 + t * 16 + l16) * K;
      v2f bv;
      bv.x = wr[k + 2 * half];
      bv.y = wr[k + 2 * half + 1];
      c[t] = wmma4(av, bv, c[t]);
    }
  }
#pragma unroll
  for (int t = 0; t < 4; ++t) {
#pragma unroll
    for (int r = 0; r < 8; ++r) {
      const int row = row0 + r + 8 * half, col = col0 + t * 16 + l16;
      float v = c[t][r];
      if (ACT == 1) v = gelu_exact(v);
      if (Cadd) v += Cadd[(size_t)row * ldo + col];
      Out[(size_t)row * ldo + col] = v;
    }
  }
}

// -------- edge bias into scores: S[b,h,n,m] = reduce_w @ edge + reduce_b --
// One workgroup per (b,n). TDM stages edge[b,:,n,:] (32x512 f32, 64KB) into
// LDS; each wave owns half the m-range and computes both 16-row h halves.
__global__ __launch_bounds__(64) void edge_bias_kernel(
    const float* __restrict__ edge, const float* __restrict__ rw,
    const float* __restrict__ rb, float* __restrict__ S) {
  __shared__ float eL[Ee * Nn];  // 64 KB
  const int bn = blockIdx.x;     // B*N
  const int b = bn >> 9, n = bn & (Nn - 1);
  const int wave = threadIdx.x >> 5, lane = threadIdx.x & 31;
  const int half = lane >> 4, l16 = lane & 15;
  if (wave == 0) {
    tdm_load_2d_f32(edge + (size_t)b * Ee * NNp + (size_t)n * Nn, eL, Nn, Ee,
                    (unsigned)NNp);
    __builtin_amdgcn_s_wait_tensorcnt(0);
  }
  __syncthreads();
  float* sb = S + (size_t)b * Hh * NNp + (size_t)n * Nn;
  for (int m0 = wave * 16; m0 < Nn; m0 += 32) {
    v8f c[2];
#pragma unroll
    for (int t = 0; t < 2; ++t)
#pragma unroll
      for (int r = 0; r < 8; ++r) c[t][r] = rb[t * 16 + r + 8 * half];
    for (int e = 0; e < Ee; e += 4) {
      v2f bv;
      bv.x = eL[(e + 2 * half) * Nn + m0 + l16];
      bv.y = eL[(e + 2 * half + 1) * Nn + m0 + l16];
#pragma unroll
      for (int t = 0; t < 2; ++t) {
        v2f a;
        a.x = rw[(t * 16 + l16) * Ee + e + 2 * half];
        a.y = rw[(t * 16 + l16) * Ee + e + 2 * half + 1];
        c[t] = wmma4(a, bv, c[t]);
      }
    }
#pragma unroll
    for (int t = 0; t < 2; ++t)
#pragma unroll
      for (int r = 0; r < 8; ++r)
        sb[(size_t)(t * 16 + r + 8 * half) * NNp + m0 + l16] = c[t][r];
  }
}

// -------------------- S += scale * Q @ K^T (per b,h) ----------------------
__global__ __launch_bounds__(32) void qk_kernel(const float* __restrict__ qkv,
                                                float* __restrict__ S) {
  const int bh = blockIdx.x;  // B*H
  const int b = bh >> 5, h = bh & 31;
  const int n0 = blockIdx.y * 16, m0 = blockIdx.z * 16;
  const int lane = threadIdx.x & 31, half = lane >> 4, l16 = lane & 15;
  float* sp = S + (size_t)bh * NNp + (size_t)n0 * Nn + m0;
  v8f c;
#pragma unroll
  for (int r = 0; r < 8; ++r) c[r] = sp[(size_t)(r + 8 * half) * Nn + l16];
  for (int d = 0; d < HDd; d += 4) {
    v2f a;
    {
      const size_t qi = (size_t)(b * Nn + n0 + l16) * (3 * Cc) + h * HDd + d + 2 * half;
      a.x = qkv[qi] * SCALE;
      a.y = qkv[qi + 1] * SCALE;
    }
    v2f bv;
    {
      const size_t ki = (size_t)(b * Nn + m0 + l16) * (3 * Cc) + Cc + h * HDd + d + 2 * half;
      bv.x = qkv[ki];
      bv.y = qkv[ki + 1];
    }
    c = wmma4(a, bv, c);
  }
#pragma unroll
  for (int r = 0; r < 8; ++r) sp[(size_t)(r + 8 * half) * Nn + l16] = c[r];
}

// ------ masked softmax rows; writes a; S overwritten with P = a + attn ----
__global__ void softmax_kernel(const unsigned char* __restrict__ mask,
                               float* __restrict__ S, float* __restrict__ Aout) {
  const int row = blockIdx.x * 4 + threadIdx.y;  // (b*H + h)*N + n
  const int b = row >> 14;                       // / (H*N)
  float* sp = S + (size_t)row * Nn;
  float* ap = Aout + (size_t)row * Nn;
  const unsigned char* mrow = mask + b * Nn;
  const int lane = threadIdx.x;
  float vals[16], ex[16];
  float mx = -INFINITY;
#pragma unroll
  for (int i = 0; i < 16; ++i) {
    const int m = lane + i * 32;
    const float v = sp[m];
    vals[i] = v;
    mx = fmaxf(mx, mrow[m] ? -INFINITY : v);
  }
  mx = wave_max(mx);
  float sum = 0.f;
#pragma unroll
  for (int i = 0; i < 16; ++i) {
    const int m = lane + i * 32;
    const float e = mrow[m] ? 0.f : expf(vals[i] - mx);
    ex[i] = e;
    sum += e;
  }
  sum = wave_sum(sum);
  const float inv = 1.0f / sum;
#pragma unroll
  for (int i = 0; i < 16; ++i) {
    const int m = lane + i * 32;
    const float a = ex[i] * inv;
    ap[m] = a;
    sp[m] = a + vals[i];  // P = a + residual
  }
}

// - per (b,n): TDM-stage P[b,:,n,:] into LDS; edge_out = expand_w @ P + eb  -
// - (WMMA from LDS), edge_new = edge + edge_out -> d_out, w-softmax -> wsum -
__global__ __launch_bounds__(64) void edge_out_kernel(
    const float* __restrict__ P, const float* __restrict__ edge,
    const float* __restrict__ ew, const float* __restrict__ ebias,
    const unsigned char* __restrict__ mask, float* __restrict__ edge_new,
    float* __restrict__ wsum) {
  __shared__ float pP[Hh * Nn];  // 64 KB (staged P slab)
  __shared__ float eo[Ee * Nn];  // 64 KB (edge_out row block)
  const int bn = blockIdx.x;
  const int b = bn >> 9, n = bn & (Nn - 1);
  const int wave = threadIdx.x >> 5, lane = threadIdx.x & 31;
  const int half = lane >> 4, l16 = lane & 15;
  if (wave == 0) {
    tdm_load_2d_f32(P + (size_t)b * Hh * NNp + (size_t)n * Nn, pP, Nn, Hh,
                    (unsigned)NNp);
    __builtin_amdgcn_s_wait_tensorcnt(0);
  }
  __syncthreads();
  const float* egb = edge + (size_t)b * Ee * NNp + (size_t)n * Nn;
  float* enb = edge_new + (size_t)b * Ee * NNp + (size_t)n * Nn;
  for (int m0 = wave * 16; m0 < Nn; m0 += 32) {
    v8f c[2];
#pragma unroll
    for (int t = 0; t < 2; ++t)
#pragma unroll
      for (int r = 0; r < 8; ++r) c[t][r] = ebias[t * 16 + r + 8 * half];
    for (int h = 0; h < Hh; h += 4) {
      v2f bv;
      bv.x = pP[(h + 2 * half) * Nn + m0 + l16];
      bv.y = pP[(h + 2 * half + 1) * Nn + m0 + l16];
#pragma unroll
      for (int t = 0; t < 2; ++t) {
        v2f a;
        a.x = ew[(t * 16 + l16) * Hh + h + 2 * half];
        a.y = ew[(t * 16 + l16) * Hh + h + 2 * half + 1];
        c[t] = wmma4(a, bv, c[t]);
      }
    }
#pragma unroll
    for (int t = 0; t < 2; ++t)
#pragma unroll
      for (int r = 0; r < 8; ++r) {
        const int e = t * 16 + r + 8 * half, m = m0 + l16;
        const float v = c[t][r];
        eo[e * Nn + m] = v;
        enb[(size_t)e * NNp + m] = egb[(size_t)e * NNp + m] + v;
      }
  }
  __syncthreads();
  const unsigned char* mrow = mask + b * Nn;
  for (int e = wave * 16; e < wave * 16 + 16; ++e) {
    float v[16];
    float mx = -INFINITY;
#pragma unroll
    for (int j = 0; j < 16; ++j) {
      const int m = lane + j * 32;
      const float t = eo[e * Nn + m];
      v[j] = t;
      mx = fmaxf(mx, mrow[m] ? -INFINITY : t);
    }
    mx = wave_max(mx);
    float s1 = 0.f, s2 = 0.f;
#pragma unroll
    for (int j = 0; j < 16; ++j) {
      const int m = lane + j * 32;
      const float exv = mrow[m] ? 0.f : expf(v[j] - mx);
      s1 += exv;
      s2 += exv * v[j];
    }
    s1 = wave_sum(s1);
    s2 = wave_sum(s2);
    if (lane == 0) wsum[(size_t)bn * Ee + e] = s2 / s1;
  }
}

// -------------------- node = a @ V (per b,h, 16 n-rows) -------------------
__global__ __launch_bounds__(32) void av_kernel(const float* __restrict__ Aa,
                                                const float* __restrict__ qkv,
                                                float* __restrict__ node) {
  const int bh = blockIdx.x;
  const int b = bh >> 5, h = bh & 31;
  const int n0 = blockIdx.y * 16;
  const int lane = threadIdx.x & 31, half = lane >> 4, l16 = lane & 15;
  v8f c;
#pragma unroll
  for (int r = 0; r < 8; ++r) c[r] = 0.0f;
  const float* ar = Aa + (size_t)bh * NNp + (size_t)n0 * Nn;
  for (int m = 0; m < Nn; m += 4) {
    v2f a;
    a.x = ar[(size_t)l16 * Nn + m + 2 * half];
    a.y = ar[(size_t)l16 * Nn + m + 2 * half + 1];
    v2f bv;
    const size_t vi = (size_t)(b * Nn + m + 2 * half) * (3 * Cc) + 2 * Cc + h * HDd + l16;
    bv.x = qkv[vi];
    bv.y = qkv[vi + 3 * Cc];
    c = wmma4(a, bv, c);
  }
#pragma unroll
  for (int r = 0; r < 8; ++r)
    node[(size_t)(b * Nn + n0 + r + 8 * half) * Cc + h * HDd + l16] = c[r];
}

// ---------------------------------------------------------------------------
extern "C" void kernel_launch(void* const* d_in, const int* in_sizes, int n_in,
                              void* d_out, int out_size, void* d_ws, size_t ws_size,
                              hipStream_t stream) {
  (void)in_sizes; (void)n_in; (void)out_size; (void)ws_size;
  const float* x       = (const float*)d_in[0];
  const float* edge    = (const float*)d_in[1];
  const unsigned char* mask = (const unsigned char*)d_in[2];
  const float* n1g = (const float*)d_in[3];
  const float* n1b = (const float*)d_in[4];
  const float* qkv_w = (const float*)d_in[5];
  const float* qkv_b = (const float*)d_in[6];
  const float* proj_w = (const float*)d_in[7];
  const float* proj_b = (const float*)d_in[8];
  const float* reduce_w = (const float*)d_in[9];
  const float* reduce_b = (const float*)d_in[10];
  const float* expand_w = (const float*)d_in[11];
  const float* expand_b = (const float*)d_in[12];
  const float* fc_w = (const float*)d_in[13];
  const float* fc_b = (const float*)d_in[14];
  const float* n2g = (const float*)d_in[15];
  const float* n2b = (const float*)d_in[16];
  const float* fc1_w = (const float*)d_in[17];
  const float* fc1_b = (const float*)d_in[18];
  const float* fc2_w = (const float*)d_in[19];
  const float* fc2_b = (const float*)d_in[20];

  const size_t BN = (size_t)Bb * Nn;            // 4096
  const size_t BNC = BN * Cc;                   // 2,097,152
  const size_t SZS = (size_t)Bb * Hh * NNp;     // 67,108,864

  float* ws = (float*)d_ws;
  float* ws_xn   = ws;                  // [B*N, C]
  float* ws_qkv  = ws_xn + BNC;         // [B*N, 3C]
  float* ws_node = ws_qkv + 3 * BNC;    // [B*N, C]
  float* ws_wsum = ws_node + BNC;       // [B*N, E]
  float* ws_tmp  = ws_wsum + BN * Ee;   // [B*N, C] (also reused as FFN hidden)
  float* ws_S    = ws_tmp + BNC;        // [B,H,N,N] scores -> P
  float* ws_A    = ws_S + SZS;          // [B,H,N,N] softmax a

  float* out_x = (float*)d_out;         // [B,N,C]
  float* out_edge = out_x + BNC;        // [B,E,N,N]

  // 1) LN1
  ln_kernel<<<BN, 256, 0, stream>>>(x, n1g, n1b, ws_xn);
  // 2) QKV = xn @ qkv_w^T + qkv_b
  gemm_awt_kernel<0><<<dim3(BN / 16, (3 * Cc) / 64), 32, 0, stream>>>(
      ws_xn, qkv_w, qkv_b, nullptr, ws_qkv, Cc, 3 * Cc);
  // 3) S = reduce_w @ edge + reduce_b  (TDM-staged edge slab per (b,n))
  edge_bias_kernel<<<BN, 64, 0, stream>>>(edge, reduce_w, reduce_b, ws_S);
  // 4) S += scale * Q K^T
  qk_kernel<<<dim3(Bb * Hh, Nn / 16, Nn / 16), 32, 0, stream>>>(ws_qkv, ws_S);
  // 5) softmax -> a ; S <- P = a + attn
  softmax_kernel<<<(Bb * Hh * Nn) / 4, dim3(32, 4), 0, stream>>>(mask, ws_S, ws_A);
  // 6) edge_out path (TDM-staged P slab): edge_new -> d_out, w_sum -> ws
  edge_out_kernel<<<BN, 64, 0, stream>>>(ws_S, edge, expand_w, expand_b, mask,
                                         out_edge, ws_wsum);
  // 7) node = a @ V
  av_kernel<<<dim3(Bb * Hh, Nn / 16), 32, 0, stream>>>(ws_A, ws_qkv, ws_node);
  // 8) tmp = node + wsum @ fc_w^T + fc_b   (K = 32)
  gemm_awt_kernel<0><<<dim3(BN / 16, Cc / 64), 32, 0, stream>>>(
      ws_wsum, fc_w, fc_b, ws_node, ws_tmp, Ee, Cc);
  // 9) x_out = x + tmp @ proj_w^T + proj_b
  gemm_awt_kernel<0><<<dim3(BN / 16, Cc / 64), 32, 0, stream>>>(
      ws_tmp, proj_w, proj_b, x, out_x, Cc, Cc);
  // 10) LN2 on x_out
  ln_kernel<<<BN, 256, 0, stream>>>(out_x, n2g, n2b, ws_xn);
  // 11) h = gelu(xn @ fc1_w^T + fc1_b)
  gemm_awt_kernel<1><<<dim3(BN / 16, Cc / 64), 32, 0, stream>>>(
      ws_xn, fc1_w, fc1_b, nullptr, ws_tmp, Cc, Cc);
  // 12) x_out += h @ fc2_w^T + fc2_b
  gemm_awt_kernel<0><<<dim3(BN / 16, Cc / 64), 32, 0, stream>>>(
      ws_tmp, fc2_w, fc2_b, out_x, out_x, Cc, Cc);
}